// ClipPeakMatcher_88957362635395
// MI455X (gfx1250) — compile-verified
//
#include <hip/hip_runtime.h>
#include <hip/hip_bf16.h>
#include <stdint.h>

// ---- problem constants (from reference) ----
constexpr int   BB    = 4;
constexpr int   NN    = 20;        // instances
constexpr int   TT    = 8;         // frames
constexpr int   HH    = 160;
constexpr int   WW    = 160;
constexpr int   PP    = HH * WW;   // 25600 reference points
constexpr int   CC    = 40;        // classes
constexpr int   PT    = PP / 1024; // 25 points per thread
constexpr float INNER = 0.5f;
constexpr float MINWH = 0.05f;
constexpr float SUP   = 1e9f;

// per-frame staged parameter block (dwords): coef[80] | lab[20] | id[20] | act[20]
constexpr int   STG   = 140;

typedef float v2f __attribute__((ext_vector_type(2)));
typedef float v4f __attribute__((ext_vector_type(4)));
typedef float v8f __attribute__((ext_vector_type(8)));

// ---------------------------------------------------------------------------
// Prep: per-batch stable argsort by mean area; emit one contiguous 140-dword
// parameter block per frame into ws:  (cx,cy,1/wx,1/wy)*20 | lab | id | act.
// ---------------------------------------------------------------------------
__global__ void ClipPeak_prep_kernel(const int* __restrict__ gl,
                                     const float* __restrict__ gb,
                                     const int* __restrict__ gid,
                                     float* __restrict__ wstage) {
    int b = threadIdx.x;
    if (b >= BB) return;

    float area[NN];
    int   ord[NN];
    int   vld[NN];
    for (int n = 0; n < NN; ++n) {
        float acc = 0.f;
        int anyv = 0;
        for (int t = 0; t < TT; ++t) {
            const float* bx = gb + (((size_t)(b * NN + n) * TT + t) * 4);
            float w = bx[2] - bx[0];
            float h = bx[3] - bx[1];
            acc += w * h;
            anyv |= (w > 0.f && h > 0.f) ? 1 : 0;
        }
        area[n] = acc / (float)TT;
        vld[n]  = anyv;
        ord[n]  = n;
    }
    // stable insertion sort ascending (matches jnp.argsort stability)
    for (int i = 1; i < NN; ++i) {
        int key = ord[i];
        float ka = area[key];
        int j = i - 1;
        while (j >= 0 && area[ord[j]] > ka) { ord[j + 1] = ord[j]; --j; }
        ord[j + 1] = key;
    }

    for (int n = 0; n < NN; ++n) {
        int m   = ord[n];
        int lab = gl[b * NN + m];
        for (int t = 0; t < TT; ++t) {
            const float* bx = gb + (((size_t)(b * NN + m) * TT + t) * 4);
            float cx = (bx[0] + bx[2]) * 0.5f;
            float cy = (bx[1] + bx[3]) * 0.5f;
            float w  = bx[2] - bx[0];
            float h  = bx[3] - bx[1];
            float iwx = 1.f / fmaxf(w, MINWH);
            float iwy = 1.f / fmaxf(h, MINWH);
            int f = b * TT + t;
            float* st = wstage + (size_t)f * STG;
            st[n * 4 + 0] = cx;
            st[n * 4 + 1] = cy;
            st[n * 4 + 2] = iwx;
            st[n * 4 + 3] = iwy;
            int idv = gid[(size_t)(b * NN + m) * TT + t];
            ((int*)st)[80 + n]  = lab;
            ((int*)st)[100 + n] = idv;
            ((int*)st)[120 + n] = (vld[m] && lab >= 0 && idv != -1) ? 1 : 0;
        }
    }
}

// ---------------------------------------------------------------------------
// Matcher: one 1024-thread block per (b,t) frame. Sequential greedy scan with
// block-wide (min,argmin) LDS reduction per instance, then streaming output.
// md is produced as a dense matmul  Score[P,20] x Onehot[20,40]  using
// V_WMMA_F32_16X16X4_F32 (K=20 as five K=4 chunks), fully hidden under the
// 131 MB non-temporal store stream (HBM floor ~6us at 23.3 TB/s).
// Frame parameters are staged LDS-side with GLOBAL_LOAD_ASYNC_TO_LDS_B32.
// ---------------------------------------------------------------------------
__global__ __launch_bounds__(1024, 1) void ClipPeak_match_kernel(
        const float* __restrict__ wstage, float* __restrict__ out) {
    __shared__ uint32_t s_claim32[PP / 4];   // claimer byte per point (0xFF = none)
    __shared__ float    s_val[1024];
    __shared__ int      s_idx[1024];
    __shared__ uint32_t s_stage[STG];        // coef[80] | lab[20] | id[20] | act[20]

    uint8_t*     s_claim = (uint8_t*)s_claim32;
    const float* s_coef  = (const float*)s_stage;
    const int*   s_lab   = (const int*)&s_stage[80];
    const int*   s_id    = (const int*)&s_stage[100];
    const int*   s_act   = (const int*)&s_stage[120];

    const int tid = threadIdx.x;
    const int f   = blockIdx.x;      // frame = b*T + t

    // ---- async-stage the 140-dword frame parameter block into LDS ----
    if (tid < STG) {
        unsigned long long gaddr =
            (unsigned long long)(uintptr_t)(wstage + (size_t)f * STG + tid);
        unsigned lds_off =
            (unsigned)(uintptr_t)(__attribute__((address_space(3))) uint32_t*)&s_stage[tid];
        asm volatile("global_load_async_to_lds_b32 %0, %1, off"
                     :: "v"(lds_off), "v"(gaddr) : "memory");
    }
    // overlap: fill the claimer array with 0xFF (none) while the async copy is in flight
    for (int i = tid; i < PP / 4; i += 1024) s_claim32[i] = 0xFFFFFFFFu;
    asm volatile("s_wait_asynccnt 0x0" ::: "memory");
    __syncthreads();

    // this thread owns 25 contiguous points
    const int pbase = tid * PT;
    float pxs[PT], pys[PT];
#pragma unroll
    for (int j = 0; j < PT; ++j) {
        int p = pbase + j;
        pxs[j] = ((float)(p % WW) + 0.5f) * (1.f / (float)WW);
        pys[j] = ((float)(p / WW) + 0.5f) * (1.f / (float)HH);
    }

    uint32_t cmask = 0;   // claimed bits for my 25 points
    float dc[PT];

    for (int n = 0; n < NN; ++n) {
        if (!s_act[n]) continue;   // uniform skip: no claims, no suppression
        const float cx  = s_coef[n * 4 + 0];
        const float cy  = s_coef[n * 4 + 1];
        const float iwx = s_coef[n * 4 + 2];
        const float iwy = s_coef[n * 4 + 3];

        float bv = 3.0e38f;
        int   bi = 0;
#pragma unroll
        for (int j = 0; j < PT; ++j) {
            float d;
            if ((cmask >> j) & 1u) {
                d = SUP;               // suppressed (claimed) point
            } else {
                float dx = (cx - pxs[j]) * iwx;
                float dy = (cy - pys[j]) * iwy;
                d = dx * dx + dy * dy;
            }
            dc[j] = d;
            if (d < bv) { bv = d; bi = pbase + j; }  // strict < keeps first index
        }

        // block-wide (min, first-index) reduction: min<0.5 <=> inner.any()
        s_val[tid] = bv;
        s_idx[tid] = bi;
        __syncthreads();
        for (int off = 512; off > 0; off >>= 1) {
            if (tid < off) {
                float v2 = s_val[tid + off]; int i2 = s_idx[tid + off];
                float v1 = s_val[tid];       int i1 = s_idx[tid];
                if (v2 < v1 || (v2 == v1 && i2 < i1)) { s_val[tid] = v2; s_idx[tid] = i2; }
            }
            __syncthreads();
        }
        float gmin = s_val[0];
        int   gidx = s_idx[0];
        __syncthreads();

        if (gmin < INNER) {
            // claim all inner points (they are necessarily unclaimed)
#pragma unroll
            for (int j = 0; j < PT; ++j) {
                if (dc[j] < INNER) {
                    cmask |= (1u << j);
                    s_claim[pbase + j] = (uint8_t)n;          // inner claim
                }
            }
        } else {
            // claim single argmin point; tag as argmin claim (score == 0)
            if (gidx >= pbase && gidx < pbase + PT) {
                cmask |= (1u << (gidx - pbase));
                s_claim[gidx] = (uint8_t)(n + 32);
            }
        }
    }
    __syncthreads();

    // ---------------- outputs (flat concat: ml | md | mi) ----------------
    float* out_ml = out + (size_t)f * PP;
    float* out_md = out + (size_t)BB * TT * PP + (size_t)f * PP * CC;
    float* out_mi = out + (size_t)BB * TT * PP + (size_t)BB * TT * PP * CC + (size_t)f * PP;

    for (int p = tid; p < PP; p += 1024) {
        int cl = s_claim[p];
        float mlv = -1.f, miv = -1.f;
        if (cl != 255) {
            int n = cl & 31;
            mlv = (float)s_lab[n];
            miv = (float)s_id[n];
        }
        __builtin_nontemporal_store(mlv, &out_ml[p]);
        __builtin_nontemporal_store(miv, &out_mi[p]);
    }

    // md[P,40] = Score[P,20] x Onehot[20,40] via V_WMMA_F32_16X16X4_F32.
    // A tile: 16 points x 4 instances (v2f/lane); B tile: 4 instances x 16
    // classes one-hot (v2f/lane); C/D: v8f. K=20 -> five K=4 chunks, no pad.
    const int wave = tid >> 5;
    const int lane = tid & 31;
    const int rrow = lane & 15;   // tile row (A) / tile col (B,D)
    const int hi   = lane >> 4;   // K-half select per ISA f32 A/B layout

    // B fragments are frame-constant: build all 3(ct) x 5(kc) once per lane.
    v2f Bt[3][5];
#pragma unroll
    for (int ct = 0; ct < 3; ++ct) {
        int c = ct * 16 + rrow;
#pragma unroll
        for (int kc = 0; kc < 5; ++kc) {
            int k0 = kc * 4 + hi * 2;
            Bt[ct][kc].x = (s_lab[k0]     == c) ? 1.f : 0.f;
            Bt[ct][kc].y = (s_lab[k0 + 1] == c) ? 1.f : 0.f;
        }
    }

    for (int pt = wave; pt < PP / 16; pt += 32) {
        int pb = pt * 16;
        // wave-uniform fast path: 16 claimer bytes all 0xFF -> md rows are zero
        bool anyClaim = false;
#pragma unroll
        for (int q = 0; q < 4; ++q)
            anyClaim |= (s_claim32[pt * 4 + q] != 0xFFFFFFFFu);
        if (!anyClaim) {
            v4f z = {0.f, 0.f, 0.f, 0.f};
            v4f* dst = (v4f*)(out_md + (size_t)pb * CC);
#pragma unroll
            for (int k = 0; k < (16 * CC) / (32 * 4); ++k)
                __builtin_nontemporal_store(z, &dst[lane + k * 32]);
            continue;
        }

        // per-lane row state: only inner claims (cl<32) contribute to md
        int p  = pb + rrow;
        int cl = s_claim[p];
        int nA = -1;
        float score = 0.f;
        if (cl < 32) {
            nA = cl;
            float px = ((float)(p % WW) + 0.5f) * (1.f / (float)WW);
            float py = ((float)(p / WW) + 0.5f) * (1.f / (float)HH);
            float dx = (s_coef[nA * 4 + 0] - px) * s_coef[nA * 4 + 2];
            float dy = (s_coef[nA * 4 + 1] - py) * s_coef[nA * 4 + 3];
            float d  = dx * dx + dy * dy;
            score = 1.f - 2.f * fminf(fmaxf(d, 0.f), INNER);
        }

        // A fragments once per tile (shared by all 3 class tiles)
        v2f At[5];
#pragma unroll
        for (int kc = 0; kc < 5; ++kc) {
            int k0 = kc * 4 + hi * 2;
            At[kc].x = (nA == k0)     ? score : 0.f;
            At[kc].y = (nA == k0 + 1) ? score : 0.f;
        }

#pragma unroll
        for (int ct = 0; ct < 3; ++ct) {       // class tiles 0..15,16..31,32..47
            int c = ct * 16 + rrow;
            v8f acc = {0.f, 0.f, 0.f, 0.f, 0.f, 0.f, 0.f, 0.f};
#pragma unroll
            for (int kc = 0; kc < 5; ++kc) {   // K = 20 instances
                acc = __builtin_amdgcn_wmma_f32_16x16x4_f32(
                        false, At[kc], false, Bt[ct][kc], (short)0, acc, false, false);
            }
            if (c < CC) {
#pragma unroll
                for (int v = 0; v < 8; ++v) {
                    int prow = pb + v + hi * 8;   // D layout: VGPR v -> row v / v+8
                    __builtin_nontemporal_store(acc[v], &out_md[(size_t)prow * CC + c]);
                }
            }
        }
    }
}

extern "C" void kernel_launch(void* const* d_in, const int* in_sizes, int n_in,
                              void* d_out, int out_size, void* d_ws, size_t ws_size,
                              hipStream_t stream) {
    (void)in_sizes; (void)n_in; (void)out_size; (void)ws_size;
    const int*   gl  = (const int*)d_in[0];    // gt_labels [B,N] int32
    const float* gb  = (const float*)d_in[1];  // gt_boxes  [B,N,T,4] f32
    const int*   gid = (const int*)d_in[2];    // gt_ids    [B,N,T] int32
    // d_in[3] ref_points / d_in[4] spatial_shapes recomputed analytically

    float* wstage = (float*)d_ws;              // B*T frames x 140 dwords (~18 KB)

    ClipPeak_prep_kernel<<<1, 32, 0, stream>>>(gl, gb, gid, wstage);
    ClipPeak_match_kernel<<<BB * TT, 1024, 0, stream>>>(wstage, (float*)d_out);
}